// Model_NPZD_68401649156380
// MI455X (gfx1250) — compile-verified
//
#include <hip/hip_runtime.h>
#include <math.h>
#include <stdint.h>

// Problem constants (match reference file)
#define B_     2048
#define WK_    52
#define HRS_   8760
#define NSTEP  56      // range_pred / dt = 7 / 0.125
#define DT_    0.125f

#define ROW    57                // padded LDS row stride (odd -> bank-conflict-free)
#define NELEM  (WK_ * NSTEP)     // 2912 compacted gather elements per array
#define LDSN   (WK_ * ROW)       // 2964 floats per LDS array

__global__ __launch_bounds__(64)
void npzd_step_kernel(const float* __restrict__ X_in,
                      const float* __restrict__ gf,
                      const float* __restrict__ gm,
                      const float* __restrict__ pvv,
                      float* __restrict__ out)
{
    __shared__ float f_lds[LDSN];
    __shared__ float m_lds[LDSN];

    const int b   = blockIdx.x;
    const int tid = threadIdx.x;

    const float* fbase = gf + (size_t)b * HRS_;
    const float* mbase = gm + (size_t)b * HRS_;

    // Stage the stride-3 gather, compacted + padded, into LDS via CDNA5
    // async global->LDS loads (tracked by ASYNCcnt).
    // Compact index j = 56*w + k  maps to  global float index 3*j  (byte 12*j)
    // and LDS float index 57*w + k = j + j/56.
    for (int j = tid; j < NELEM; j += 64) {
        unsigned dstb = (unsigned)(j + j / 56) * 4u;      // padded byte offset in LDS
        unsigned ldsf = (unsigned)(uintptr_t)f_lds + dstb;
        unsigned ldsm = (unsigned)(uintptr_t)m_lds + dstb;
        unsigned voff = 12u * (unsigned)j;                // global byte offset
        asm volatile("global_load_async_to_lds_b32 %0, %1, %2"
                     :: "v"(ldsf), "v"(voff), "s"(fbase) : "memory");
        asm volatile("global_load_async_to_lds_b32 %0, %1, %2"
                     :: "v"(ldsm), "v"(voff), "s"(mbase) : "memory");
    }
    asm volatile("s_wait_asynccnt 0" ::: "memory");
    __syncthreads();

    const int w = tid;
    if (w >= WK_) return;

    // Per-batch parameters fused with the fixed pv[] constants (block-uniform).
    const float* p   = pvv + b * 10;
    const float chi  = p[0] * 1.0f;
    const float rho2 = p[1] * 2.0f;
    const float gam  = p[2] * 0.1f;
    const float nlam = p[3] * -0.05f;   // sign folded: exp(nlam * Pc)
    const float eps  = p[4] * 0.1f;
    const float alp  = p[5] * 0.3f;
    const float bet  = p[6] * 0.6f;
    const float eta  = p[7] * 0.15f;
    const float phi  = p[8] * 0.4f;
    const float zet  = p[9] * 0.1f;
    const float rem  = 1.0f - alp - bet;  // (1 - alp*pv5 - bet*pv6)

    // Initial state from X_in[b, w, 1..4, 0]
    const size_t xb = ((size_t)b * WK_ + w) * 5;
    float N = X_in[xb + 1];
    float P = X_in[xb + 2];
    float Z = X_in[xb + 3];
    float D = X_in[xb + 4];

    // Output layout: (B, WK, 4, 8)
    float* ob = out + ((size_t)b * WK_ + w) * 32;
    ob[0]  = N;   // sample s=0
    ob[8]  = P;
    ob[16] = Z;
    ob[24] = D;

    const float thr = 0.01f;
    const float Q0  = 8.0f;
    const float dt  = DT_;
    const int  base = ROW * w;

    #pragma unroll 8
    for (int k = 0; k < NSTEP; ++k) {
        float ft = f_lds[base + k];
        float mt = m_lds[base + k];

        float Pc = fmaxf(thr, P);
        float Zc = fmaxf(thr, Z);
        float gN = N / (chi + N);
        float zg = rho2 * (1.0f - __expf(nlam * Pc)) * Zc;
        float up = gN * ft * Pc;                       // Vm = 1.0

        float Nn = N + dt * (-up + alp * zg + eps * P + gam * Z + phi * D + mt * (Q0 - N));
        float Pn = P + dt * (up - zg - eps * P - eta * P - mt * P);
        float Zn = Z + dt * (bet * zg - gam * Z - mt * Z);
        float Dn = D + dt * (eta * P + rem * zg - phi * D - zet * D - mt * D);
        N = Nn; P = Pn; Z = Zn; D = Dn;

        int kp = k + 1;
        if ((kp & 7) == 0) {
            int s = kp >> 3;
            ob[s]      = N;
            ob[8 + s]  = P;
            ob[16 + s] = Z;
            ob[24 + s] = D;
        }
    }
}

extern "C" void kernel_launch(void* const* d_in, const int* in_sizes, int n_in,
                              void* d_out, int out_size, void* d_ws, size_t ws_size,
                              hipStream_t stream) {
    const float* X_in = (const float*)d_in[0];
    const float* gf   = (const float*)d_in[1];
    const float* gm   = (const float*)d_in[2];
    const float* pv   = (const float*)d_in[3];
    float* out = (float*)d_out;
    (void)in_sizes; (void)n_in; (void)out_size; (void)d_ws; (void)ws_size;

    npzd_step_kernel<<<B_, 64, 0, stream>>>(X_in, gf, gm, pv, out);
}